// RegressionModel_65481071410619
// MI455X (gfx1250) — compile-verified
//
#include <hip/hip_runtime.h>

// ---------------------------------------------------------------------------
// 14-qubit variational circuit simulator for MI455X (gfx1250, wave32, WGP).
//   - One workgroup per batch element; state vector ping-pongs in 256 KB LDS.
//   - Each rotation layer is fused into 3x (16x16 complex group unitary,
//     applied with v_wmma_f32_16x16x4_f32) + 1x (4x4 VALU group).
//   - Each CNOT ring is a GF(2)-linear index permutation (one LDS scatter).
// ---------------------------------------------------------------------------

typedef float v2f __attribute__((ext_vector_type(2)));
typedef float v8f __attribute__((ext_vector_type(8)));

#define NW    14
#define AMPS  16384          // 2^14 amplitudes
#define TPB   256            // 8 wave32 waves
#define WAVES (TPB / 32)

// D(16x16,f32) = A(16x4,f32) * B(4x16,f32) + C ; signature mirrored from the
// probe-confirmed f16 variant: (neg_a, A, neg_b, B, c_mod, C, reuse_a, reuse_b)
__device__ __forceinline__ v8f wmma_k4(v2f a, v2f b, v8f c) {
  return __builtin_amdgcn_wmma_f32_16x16x4_f32(false, a, false, b, (short)0, c,
                                               false, false);
}

__global__ __launch_bounds__(TPB)
void qsel_circuit_kernel(const float* __restrict__ state_batch,
                         const float* __restrict__ params,
                         const float* __restrict__ head_w,
                         const float* __restrict__ head_b,
                         float* __restrict__ out)
{
  extern __shared__ float smem[];
  float* bufR0 = smem;                  // 16384
  float* bufI0 = smem + AMPS;           // 16384
  float* bufR1 = smem + 2 * AMPS;       // 16384
  float* bufI1 = smem + 3 * AMPS;       // 16384
  float* sUr   = smem + 4 * AMPS;       // 14*4  one-qubit gate table (re)
  float* sUi   = sUr + NW * 4;          // 14*4  (im)
  float* gR    = sUi + NW * 4;          // 16*17 group unitary Re (padded)
  float* gI    = gR + 16 * 17;          // 16*17 group unitary Im
  float* gN    = gI + 16 * 17;          // 16*17 -Im (F32 WMMA has no A-neg)
  float* cLo   = gN + 16 * 17;          // 128   measurement coeff (low bits)
  float* cHi   = cLo + 128;             // 128   measurement coeff (high bits)

  const int tid  = threadIdx.x;
  const int lane = tid & 31;
  const int wv   = tid >> 5;
  const int blk  = blockIdx.x;

  const float* X  = state_batch + (size_t)blk * (2 * AMPS); // only signs used
  const float* Pp = params + (size_t)blk * (2 * NW * 3);

  float* curR = bufR0; float* curI = bufI0;
  float* nxtR = bufR1; float* nxtI = bufI1;

  // psi = |0...0>
  for (int i = tid; i < AMPS; i += TPB) {
    curR[i] = (i == 0) ? 1.f : 0.f;
    curI[i] = 0.f;
  }
  __syncthreads();

  // 7 rotation layers: ly=0 AngleEmbedding(RX), ly=1..6 Rot layers (SEL),
  // weights shared across the 3 blocks; CNOT ring after each Rot layer.
  for (int ly = 0; ly < 7; ++ly) {
    // ---- build the 14 one-qubit 2x2 gates of this layer -------------------
    if (tid < NW) {
      int w = tid;
      float u00r, u00i, u01r, u01i, u10r, u10i, u11r, u11i;
      if (ly == 0) {
        // torch.angle(real) -> 0 or pi ; RX(0)=I, RX(pi)=[[0,-i],[-i,0]]
        float cc = (X[w] < 0.f) ? 0.f : 1.f;
        float ss = 1.f - cc;
        u00r = cc;  u00i = 0.f;  u01r = 0.f; u01i = -ss;
        u10r = 0.f; u10i = -ss;  u11r = cc;  u11i = 0.f;
      } else {
        int l = (ly - 1) & 1;                       // layer index within SEL
        const float* pw = Pp + (l * NW + w) * 3;
        float phi = pw[0], th = pw[1], om = pw[2];
        float ch = cosf(0.5f * th), sh = sinf(0.5f * th);
        float a = 0.5f * (phi + om), d = 0.5f * (phi - om);
        float ca = cosf(a), sa = sinf(a);
        float cd = cosf(d), sd = sinf(d);
        u00r =  ca * ch; u00i = -sa * ch;           // e^{-i(phi+om)/2} c
        u01r = -cd * sh; u01i = -sd * sh;           // -e^{+i(phi-om)/2} s
        u10r =  cd * sh; u10i = -sd * sh;           // e^{-i(phi-om)/2} s
        u11r =  ca * ch; u11i =  sa * ch;           // e^{+i(phi+om)/2} c
      }
      sUr[w * 4 + 0] = u00r; sUi[w * 4 + 0] = u00i;
      sUr[w * 4 + 1] = u01r; sUi[w * 4 + 1] = u01i;
      sUr[w * 4 + 2] = u10r; sUi[w * 4 + 2] = u10i;
      sUr[w * 4 + 3] = u11r; sUi[w * 4 + 3] = u11i;
    }
    __syncthreads();

    // ---- three fused 4-wire groups via WMMA f32 16x16x4 -------------------
    for (int g = 0; g < 3; ++g) {
      const int p = g << 2;                 // amp bit offset of this group
      {                                     // G = kron of 4 gates (16x16)
        int i = tid >> 4, k = tid & 15;
        float cr = 1.f, ci = 0.f;
        for (int j = 0; j < 4; ++j) {
          int w = 13 - (p + j);             // wire acting on amp bit p+j
          int e = ((i >> j) & 1) * 2 + ((k >> j) & 1);
          float ur = sUr[w * 4 + e], ui = sUi[w * 4 + e];
          float nr = cr * ur - ci * ui;
          float ni = cr * ui + ci * ur;
          cr = nr; ci = ni;
        }
        gR[i * 17 + k] = cr; gI[i * 17 + k] = ci; gN[i * 17 + k] = -ci;
      }
      __syncthreads();

      const int S    = 1 << p;
      const int lm   = S - 1;
      const int n    = lane & 15;           // tile column / A-matrix row
      const int half = lane >> 4;
      for (int t = wv; t < 64; t += WAVES) {          // 64 tiles of 16 strings
        int bid  = t * 16 + n;
        int base = ((bid >> p) << (p + 4)) | (bid & lm);
        v8f accR = {0.f, 0.f, 0.f, 0.f, 0.f, 0.f, 0.f, 0.f};
        v8f accI = {0.f, 0.f, 0.f, 0.f, 0.f, 0.f, 0.f, 0.f};
        for (int c = 0; c < 4; ++c) {       // K = 16 as 4 chained K=4 WMMAs
          int kk = (c << 2) + (half << 1);
          v2f bre = { curR[base + kk * S], curR[base + (kk + 1) * S] };
          v2f bim = { curI[base + kk * S], curI[base + (kk + 1) * S] };
          v2f are = { gR[n * 17 + kk], gR[n * 17 + kk + 1] };
          v2f aim = { gI[n * 17 + kk], gI[n * 17 + kk + 1] };
          v2f ani = { gN[n * 17 + kk], gN[n * 17 + kk + 1] };
          accR = wmma_k4(are, bre, accR);   // += Re(G)*Re(S)
          accR = wmma_k4(ani, bim, accR);   // -= Im(G)*Im(S)
          accI = wmma_k4(are, bim, accI);   // += Re(G)*Im(S)
          accI = wmma_k4(aim, bre, accI);   // += Im(G)*Re(S)
        }
        for (int v = 0; v < 8; ++v) {       // D row M = v + 8*half
          int i = v + (half << 3);
          nxtR[base + i * S] = accR[v];
          nxtI[base + i * S] = accI[v];
        }
      }
      __syncthreads();
      float* tR = curR; curR = nxtR; nxtR = tR;
      float* tI = curI; curI = nxtI; nxtI = tI;
    }

    // ---- top 2-wire group (amp bits 12..13 = wires 1,0): 4x4 via VALU -----
    {
      float tr[16], ti[16];
      for (int i = 0; i < 4; ++i)
        for (int k = 0; k < 4; ++k) {
          int e1 = (i & 1) * 2 + (k & 1);                   // wire 1 (bit 12)
          int e0 = ((i >> 1) & 1) * 2 + ((k >> 1) & 1);     // wire 0 (bit 13)
          float ar = sUr[1 * 4 + e1], ai = sUi[1 * 4 + e1];
          float br = sUr[0 * 4 + e0], bi = sUi[0 * 4 + e0];
          tr[i * 4 + k] = ar * br - ai * bi;
          ti[i * 4 + k] = ar * bi + ai * br;
        }
      for (int b = tid; b < 4096; b += TPB) {
        float xr[4], xi[4];
        for (int k = 0; k < 4; ++k) {
          xr[k] = curR[b + k * 4096];
          xi[k] = curI[b + k * 4096];
        }
        for (int i = 0; i < 4; ++i) {
          float yr = 0.f, yi = 0.f;
          for (int k = 0; k < 4; ++k) {
            yr += tr[i * 4 + k] * xr[k] - ti[i * 4 + k] * xi[k];
            yi += tr[i * 4 + k] * xi[k] + ti[i * 4 + k] * xr[k];
          }
          nxtR[b + i * 4096] = yr;
          nxtI[b + i * 4096] = yi;
        }
      }
      __syncthreads();
      float* tR = curR; curR = nxtR; nxtR = tR;
      float* tI = curI; curI = nxtI; nxtI = tI;
    }

    // ---- CNOT ring as composed GF(2)-linear permutation (Rot layers only) -
    if (ly > 0) {
      int r = ((ly - 1) & 1) + 1;           // PennyLane ranges: l%13 + 1
      unsigned msk[NW];
      for (int b = 0; b < NW; ++b) msk[b] = 1u << b;
      for (int w = 0; w < NW; ++w) {        // bit_t ^= bit_c, sequentially
        int cb = 13 - w;
        int tb = 13 - ((w + r) % NW);
        msk[tb] ^= msk[cb];
      }
      for (int idx = tid; idx < AMPS; idx += TPB) {
        unsigned j = 0;
        for (int b = 0; b < NW; ++b)
          j |= (unsigned)(__builtin_popcount((unsigned)idx & msk[b]) & 1) << b;
        nxtR[j] = curR[idx];                // psi'[P(x)] = psi[x]
        nxtI[j] = curI[idx];
      }
      __syncthreads();
      float* tR = curR; curR = nxtR; nxtR = tR;
      float* tI = curI; curI = nxtI; nxtI = tI;
    }
  }

  // ---- measurement: sum |psi|^2 * sum_w head_w[w]*(1-2*bit_{13-w}) --------
  if (tid < 128) {
    float s = 0.f;
    for (int b = 0; b < 7; ++b)
      s += head_w[13 - b] * (1.f - 2.f * (float)((tid >> b) & 1));
    cLo[tid] = s;
  } else {
    int m = tid - 128;
    float s = 0.f;
    for (int b = 0; b < 7; ++b)
      s += head_w[6 - b] * (1.f - 2.f * (float)((m >> b) & 1));
    cHi[m] = s;
  }
  __syncthreads();

  float local = 0.f;
  for (int idx = tid; idx < AMPS; idx += TPB) {
    float re = curR[idx], im = curI[idx];
    local += (re * re + im * im) * (cLo[idx & 127] + cHi[idx >> 7]);
  }
  nxtR[tid] = local;                        // spare buffer as reduction pad
  __syncthreads();
  for (int s = TPB / 2; s > 0; s >>= 1) {
    if (tid < s) nxtR[tid] += nxtR[tid + s];
    __syncthreads();
  }
  if (tid == 0) out[blk] = nxtR[0] + head_b[0];
}

extern "C" void kernel_launch(void* const* d_in, const int* in_sizes, int n_in,
                              void* d_out, int out_size, void* d_ws,
                              size_t ws_size, hipStream_t stream) {
  const float* state  = (const float*)d_in[0];  // (B, 2^14, 2) f32
  const float* params = (const float*)d_in[1];  // (B, 84) f32
  const float* hw     = (const float*)d_in[2];  // (1, 14) f32
  const float* hb     = (const float*)d_in[3];  // (1,) f32
  float* out = (float*)d_out;

  const int B = in_sizes[1] / (2 * NW * 3);     // 512

  const size_t shmem =
      (size_t)(4 * AMPS + 2 * NW * 4 + 3 * 16 * 17 + 2 * 128) * sizeof(float);

  // 320 KB LDS per WGP on CDNA5; opt this kernel into >48 KB dynamic LDS.
  (void)hipFuncSetAttribute((const void*)qsel_circuit_kernel,
                            hipFuncAttributeMaxDynamicSharedMemorySize,
                            (int)shmem);

  qsel_circuit_kernel<<<dim3(B), dim3(TPB), shmem, stream>>>(state, params, hw,
                                                             hb, out);
}